// SimpleArchitecture_44555990728725
// MI455X (gfx1250) — compile-verified
//
#include <hip/hip_runtime.h>

typedef float v2f __attribute__((ext_vector_type(2)));
typedef float v8f __attribute__((ext_vector_type(8)));

// ---------------------------------------------------------------------------
// Zero-fill
// ---------------------------------------------------------------------------
__global__ void zero_f(float* __restrict__ p, int n) {
  int i = blockIdx.x * blockDim.x + threadIdx.x;
  if (i < n) p[i] = 0.0f;
}

// ---------------------------------------------------------------------------
// Y[nrows x OUTC] = X[nrows x INC] @ W[OUTC x INC]^T + bias[OUTC]
// fp32 WMMA 16x16x4. Each wave computes MT row-tiles (16 rows each) so the
// B (weight) fragments are loaded once per k-step and reused across MT tiles.
//
// A (16x4 f32): lane L holds row M = L&15, K-pair base = (L>>4)*2.
// B (4x16 f32): lane L holds col N = L&15, same K-pair base (B[k][n] = W[n][k]).
// C/D (16x16 f32): lane L covers col N = L&15, rows M = (L>>4)*8 + v, v=0..7.
// ---------------------------------------------------------------------------
template <int INC, int OUTC, int MT>
__global__ void wmma_linear(const float* __restrict__ X,
                            const float* __restrict__ W,
                            const float* __restrict__ Bias,
                            float* __restrict__ Y, int nrows) {
  constexpr int NT = OUTC / 16;
  int wave = (int)((blockIdx.x * blockDim.x + threadIdx.x) >> 5);
  int lane = threadIdx.x & 31;
  int row0 = wave * 16 * MT;
  if (row0 >= nrows) return;  // wave-uniform exit; nrows % (16*MT) == 0

  int nm = lane & 15;          // row (for A) == col (for B, C/D) index
  int kg = (lane >> 4) << 1;   // K-pair base: 0 or 2

  const float* xrow[MT];
#pragma unroll
  for (int mt = 0; mt < MT; ++mt) {
    xrow[mt] = X + (size_t)(row0 + mt * 16 + nm) * INC;
    __builtin_prefetch(xrow[mt], 0, 3);
  }
  const float* wrow[NT];
#pragma unroll
  for (int t = 0; t < NT; ++t) wrow[t] = W + (size_t)(t * 16 + nm) * INC;

  v8f acc[MT][NT];
#pragma unroll
  for (int t = 0; t < NT; ++t) {
    float bv = Bias[t * 16 + nm];
#pragma unroll
    for (int mt = 0; mt < MT; ++mt)
#pragma unroll
      for (int v = 0; v < 8; ++v) acc[mt][t][v] = bv;
  }

#pragma unroll 4
  for (int k = 0; k < INC; k += 4) {
    v2f b[NT];
#pragma unroll
    for (int t = 0; t < NT; ++t) {
      b[t].x = wrow[t][k + kg];
      b[t].y = wrow[t][k + kg + 1];
    }
#pragma unroll
    for (int mt = 0; mt < MT; ++mt) {
      v2f a;
      a.x = xrow[mt][k + kg];
      a.y = xrow[mt][k + kg + 1];
#pragma unroll
      for (int t = 0; t < NT; ++t) {
        acc[mt][t] = __builtin_amdgcn_wmma_f32_16x16x4_f32(
            /*neg_a=*/false, a, /*neg_b=*/false, b[t],
            /*c_mod=*/(short)0, acc[mt][t], /*reuse_a=*/false,
            /*reuse_b=*/false);
      }
    }
  }

  int mb = (lane >> 4) * 8;
#pragma unroll
  for (int mt = 0; mt < MT; ++mt)
#pragma unroll
    for (int t = 0; t < NT; ++t)
#pragma unroll
      for (int v = 0; v < 8; ++v)
        Y[(size_t)(row0 + mt * 16 + mb + v) * OUTC + t * 16 + nm] =
            acc[mt][t][v];
}

// ---------------------------------------------------------------------------
// agg[dst[e]*C + c] += H[src[e]*C + c]   (segment_sum over edges)
// One lane per (edge, channel): gather reads coalesced, L2-resident atomics.
// ---------------------------------------------------------------------------
template <int C>
__global__ void scatter_add(const float* __restrict__ H,
                            const int* __restrict__ srcI,
                            const int* __restrict__ dstI,
                            float* __restrict__ agg, int nedges) {
  long long tid = (long long)blockIdx.x * blockDim.x + threadIdx.x;
  int e = (int)(tid / C);
  int c = (int)(tid % C);
  if (e >= nedges) return;
  int s = srcI[e];
  int d = dstI[e];
  float val = H[(size_t)s * C + c];
  __hip_atomic_fetch_add(&agg[(size_t)d * C + c], val, __ATOMIC_RELAXED,
                         __HIP_MEMORY_SCOPE_AGENT);
}

// ---------------------------------------------------------------------------
// FZ[i] = relu(FZ[i] + A[i])   (in-place combine)
// ---------------------------------------------------------------------------
__global__ void add_relu(const float* __restrict__ A, float* __restrict__ FZ,
                         int n) {
  int i = blockIdx.x * blockDim.x + threadIdx.x;
  if (i < n) {
    float v = FZ[i] + A[i];
    FZ[i] = v > 0.0f ? v : 0.0f;
  }
}

// ---------------------------------------------------------------------------
// O[r,:] = log_softmax(F[r,:] + A[r,:]) over 16 channels, 1 thread/row
// ---------------------------------------------------------------------------
__global__ void add_logsoftmax16(const float* __restrict__ F,
                                 const float* __restrict__ A,
                                 float* __restrict__ O, int nrows) {
  int r = blockIdx.x * blockDim.x + threadIdx.x;
  if (r >= nrows) return;
  float v[16];
  float mx = -3.402823466e38f;
#pragma unroll
  for (int c = 0; c < 16; ++c) {
    v[c] = F[(size_t)r * 16 + c] + A[(size_t)r * 16 + c];
    mx = fmaxf(mx, v[c]);
  }
  float s = 0.0f;
#pragma unroll
  for (int c = 0; c < 16; ++c) s += __expf(v[c] - mx);
  float l = mx + __logf(s);
#pragma unroll
  for (int c = 0; c < 16; ++c) O[(size_t)r * 16 + c] = v[c] - l;
}

// ---------------------------------------------------------------------------
// Host launcher
// ---------------------------------------------------------------------------
static inline int cdiv_ll(long long a, long long b) {
  return (int)((a + b - 1) / b);
}

extern "C" void kernel_launch(void* const* d_in, const int* in_sizes, int n_in,
                              void* d_out, int out_size, void* d_ws,
                              size_t ws_size, hipStream_t stream) {
  (void)n_in; (void)out_size; (void)ws_size;
  const float* x     = (const float*)d_in[0];
  const float* w1a_w = (const float*)d_in[1];
  const float* w1a_b = (const float*)d_in[2];
  const float* w1b_w = (const float*)d_in[3];
  const float* w1b_b = (const float*)d_in[4];
  const float* w2a_w = (const float*)d_in[5];
  const float* w2a_b = (const float*)d_in[6];
  const float* w2b_w = (const float*)d_in[7];
  const float* w2b_b = (const float*)d_in[8];
  const int*   edge  = (const int*)d_in[9];

  const int N = in_sizes[0] / 256;   // 100000 (divisible by 32)
  const int E = in_sizes[9] / 2;     // 1600000
  const int* srcI = edge;
  const int* dstI = edge + E;

  // workspace aliasing: 3 buffers of N*32 floats (38.4 MB)
  float* buf0 = (float*)d_ws;                    // h1, later h2
  float* buf1 = buf0 + (size_t)N * 32;           // agg1, later agg2
  float* buf2 = buf1 + (size_t)N * 32;           // f1/z1, later f2

  const int BLK = 256;                 // 8 waves per block
  const int MT = 2;                    // 32 rows per wave
  const int waves = N / (16 * MT);     // 3125
  const int gemm_blocks = cdiv_ll(waves, BLK / 32);

  // ---- layer 1 ----
  zero_f<<<cdiv_ll((long long)N * 32, BLK), BLK, 0, stream>>>(buf1, N * 32);
  wmma_linear<256, 32, MT><<<gemm_blocks, BLK, 0, stream>>>(x, w1a_w, w1a_b,
                                                            buf0, N);
  scatter_add<32><<<cdiv_ll((long long)E * 32, BLK), BLK, 0, stream>>>(
      buf0, srcI, dstI, buf1, E);
  wmma_linear<32, 32, MT><<<gemm_blocks, BLK, 0, stream>>>(buf0, w1b_w, w1b_b,
                                                           buf2, N);
  add_relu<<<cdiv_ll((long long)N * 32, BLK), BLK, 0, stream>>>(buf1, buf2,
                                                                N * 32);

  // ---- layer 2 ----
  zero_f<<<cdiv_ll((long long)N * 16, BLK), BLK, 0, stream>>>(buf1, N * 16);
  wmma_linear<32, 16, MT><<<gemm_blocks, BLK, 0, stream>>>(buf2, w2a_w, w2a_b,
                                                           buf0, N);
  scatter_add<16><<<cdiv_ll((long long)E * 16, BLK), BLK, 0, stream>>>(
      buf0, srcI, dstI, buf1, E);
  wmma_linear<16, 16, MT><<<gemm_blocks, BLK, 0, stream>>>(buf0, w2b_w, w2b_b,
                                                           buf2, N);
  add_logsoftmax16<<<cdiv_ll(N, BLK), BLK, 0, stream>>>(buf2, buf1,
                                                        (float*)d_out, N);
}